// RNAGraphAutoencoder_5927054868534
// MI455X (gfx1250) — compile-verified
//
#include <hip/hip_runtime.h>
#include <math.h>

// ---------------------------------------------------------------------------
// GCN autoencoder for MI455X (gfx1250, wave32).
//   * GEMMs: V_WMMA_F32_16X16X4_F32, fully unrolled, guard-free (padded dims).
//   * Message passing: float4 gathers + f32 global atomics, all L2-resident
//     (~13 MB tables vs 192 MB L2); this dominates runtime, so fp32 is free.
// ---------------------------------------------------------------------------

typedef float v2f __attribute__((ext_vector_type(2)));
typedef float v4f __attribute__((ext_vector_type(4)));
typedef float v8f __attribute__((ext_vector_type(8)));

#define TPB 256
static inline unsigned cdiv(long long a, long long b) { return (unsigned)((a + b - 1) / b); }

// ----------------------------- small kernels -------------------------------

__global__ void k_init_deg(float* __restrict__ deg, int n) {
    int i = blockIdx.x * blockDim.x + threadIdx.x;
    if (i < n) deg[i] = 1.0f;                      // self-loop contributes 1
}

__global__ void k_deg_accum(const int* __restrict__ dst, float* __restrict__ deg, int E) {
    int e = blockIdx.x * blockDim.x + threadIdx.x;
    if (e < E) atomicAdd(&deg[dst[e]], 1.0f);
}

__global__ void k_deg_final(const float* __restrict__ deg,
                            float* __restrict__ dis, float* __restrict__ dinv, int n) {
    int i = blockIdx.x * blockDim.x + threadIdx.x;
    if (i < n) {
        float d = deg[i];
        dis[i]  = rsqrtf(d);
        dinv[i] = 1.0f / d;
    }
}

// norm[e] = dis[src[e]] * dis[dst[e]]  (computed once, reused by 2..16 threads)
__global__ void k_edge_norm(const int* __restrict__ src, const int* __restrict__ dst,
                            const float* __restrict__ dis, float* __restrict__ norm, int E) {
    int e = blockIdx.x * blockDim.x + threadIdx.x;
    if (e < E) norm[e] = dis[src[e]] * dis[dst[e]];
}

__global__ void k_zero(float* __restrict__ p, long long n) {
    long long i = (long long)blockIdx.x * blockDim.x + threadIdx.x;
    if (i < n) p[i] = 0.0f;
}

// Pad x (n x 7) into (n16 x 8), zero-filling column 7 and rows >= n.
__global__ void k_pad_x(const float* __restrict__ x, float* __restrict__ xp,
                        int n, int n16) {
    long long i = (long long)blockIdx.x * blockDim.x + threadIdx.x;
    long long total = (long long)n16 * 8;
    if (i >= total) return;
    int r = (int)(i >> 3);
    int c = (int)(i & 7);
    xp[i] = (r < n && c < 7) ? x[(long long)r * 7 + c] : 0.0f;
}

// Pad W (fin x fout) into (finp x ldout), zero-filled.
__global__ void k_pad_w(const float* __restrict__ w, float* __restrict__ wp,
                        int fin, int fout, int finp, int ldout) {
    int i = blockIdx.x * blockDim.x + threadIdx.x;
    int total = finp * ldout;
    if (i >= total) return;
    int r = i / ldout;
    int c = i - r * ldout;
    wp[i] = (r < fin && c < fout) ? w[r * fout + c] : 0.0f;
}

// ------------------------- fp32 WMMA tiled GEMM ----------------------------
// C[N16, COLT*16] = A[N16, FINP] @ W[FINP, COLT*16]; everything padded, so
// no guards anywhere.  One wave32 per 16-row slab; the A fragment (float2,
// global_load_b64) is reused across COLT column tiles -> COLT*(FINP/4)
// back-to-back v_wmma_f32_16x16x4_f32 per wave.
//
// A-frag (16x4): lanes 0-15 row M=lane K=(k0,k0+1); lanes 16-31 row M=lane-16
// K=(k0+2,k0+3).  B-frag mirrored.  C/D: VGPR r = row r (lanes 0-15) /
// r+8 (lanes 16-31), col = lane&15.

template <int FINP, int COLT>
__global__ void k_gemm_wmma_f32(const float* __restrict__ A,
                                const float* __restrict__ W,
                                float* __restrict__ C,
                                int nRowTiles) {
    constexpr int LDC = COLT * 16;
    const int wave = threadIdx.x >> 5;             // wave32
    const int lane = threadIdx.x & 31;
    const int tile = blockIdx.x * (blockDim.x >> 5) + wave;
    if (tile >= nRowTiles) return;                 // wave-uniform: EXEC stays all-1s

    const int rowBase = tile * 16;
    const int half = lane >> 4;                    // 0: K pair (0,1); 1: (2,3)
    const int l15  = lane & 15;
    const int arow = rowBase + l15;

    v8f acc[COLT];
#pragma unroll
    for (int ct = 0; ct < COLT; ++ct) acc[ct] = (v8f){0.f,0.f,0.f,0.f,0.f,0.f,0.f,0.f};

#pragma unroll
    for (int k0 = 0; k0 < FINP; k0 += 4) {
        const int ka = k0 + half * 2;
        const v2f a = *(const v2f*)(A + (long long)arow * FINP + ka);
#pragma unroll
        for (int ct = 0; ct < COLT; ++ct) {
            v2f b;
            b.x = W[(ka)     * LDC + ct * 16 + l15];
            b.y = W[(ka + 1) * LDC + ct * 16 + l15];
            acc[ct] = __builtin_amdgcn_wmma_f32_16x16x4_f32(
                false, a, false, b, (short)0, acc[ct], false, false);
        }
    }

#pragma unroll
    for (int ct = 0; ct < COLT; ++ct) {
        const int col = ct * 16 + l15;
#pragma unroll
        for (int r = 0; r < 8; ++r) {
            const int row = rowBase + r + half * 8;
            C[(long long)row * LDC + col] = acc[ct][r];
        }
    }
}

// ---------------------------- edge scatter ---------------------------------
// chunks is a power of two with 4*chunks <= ld, so no per-feature guards:
// one aligned float4 gather (global_load_b128) + 4 f32 atomics per thread.

__global__ void k_edge_scatter(const float* __restrict__ xw,
                               const float* __restrict__ norm,
                               const int* __restrict__ src,
                               const int* __restrict__ dst,
                               float* __restrict__ agg,
                               int E, int ld, int chunkShift, int chunkMask) {
    long long gid = (long long)blockIdx.x * blockDim.x + threadIdx.x;
    int e = (int)(gid >> chunkShift);
    if (e >= E) return;
    int c   = (int)gid & chunkMask;
    int s   = src[e];
    int d   = dst[e];
    float w = norm[e];
    int f0  = c * 4;
    const v4f m = *(const v4f*)(xw + (long long)s * ld + f0);
    float* ap = agg + (long long)d * ld + f0;
    atomicAdd(ap + 0, w * m.x);
    atomicAdd(ap + 1, w * m.y);
    atomicAdd(ap + 2, w * m.z);
    atomicAdd(ap + 3, w * m.w);
}

// ------------------------------ epilogue -----------------------------------
// out[node, f<nfeat] = relu(agg + deg_inv*xw + b); optional mirror (latent).

__global__ void k_finalize(const float* __restrict__ agg,
                           const float* __restrict__ xw,
                           const float* __restrict__ dinv,
                           const float* __restrict__ bias,
                           float* __restrict__ out,
                           float* __restrict__ out2,
                           int n, int nfeat, int ld_in, int ld_out) {
    long long i = (long long)blockIdx.x * blockDim.x + threadIdx.x;
    long long total = (long long)n * nfeat;
    if (i >= total) return;
    int node = (int)(i / nfeat);
    int f    = (int)(i - (long long)node * nfeat);
    long long ii = (long long)node * ld_in + f;
    float v = agg[ii] + dinv[node] * xw[ii] + bias[f];
    v = v > 0.0f ? v : 0.0f;
    long long oi = (long long)node * ld_out + f;
    out[oi] = v;
    if (out2) out2[oi] = v;
}

// ------------------------------- driver ------------------------------------

extern "C" void kernel_launch(void* const* d_in, const int* in_sizes, int n_in,
                              void* d_out, int out_size, void* d_ws, size_t ws_size,
                              hipStream_t stream) {
    const float* x  = (const float*)d_in[0];
    const int*   ei = (const int*)d_in[1];
    const int N   = in_sizes[0] / 7;        // INPUT_DIM = 7
    const int E   = in_sizes[1] / 2;        // edge_index is [2, E]
    const int N16 = ((N + 15) / 16) * 16;
    const int* src = ei;
    const int* dst = ei + E;

    const float* Wt[6];
    const float* bt[6];
    for (int i = 0; i < 6; ++i) {
        Wt[i] = (const float*)d_in[2 + 2 * i];
        bt[i] = (const float*)d_in[3 + 2 * i];
    }

    float* out    = (float*)d_out;
    float* latent = out + (long long)N * 7;  // (h, latent) concatenated flat

    // Workspace layout (fp32, 64B-aligned slabs):
    // deg[N] | dis[N] | dinv[N] | norm[E] | xpad[N16*8] | w0p[8*64] | w5p[64*16]
    // | xw[N16*64] | agg[N16*64] | hbuf[N16*64]
    float* deg  = (float*)d_ws;
    float* dis  = deg  + N16;
    float* dinv = dis  + N16;
    float* norm = dinv + N16;
    float* xpad = norm + ((E + 15) & ~15);
    float* w0p  = xpad + (long long)N16 * 8;
    float* w5p  = w0p  + 8 * 64;
    float* xw   = w5p  + 64 * 16;
    float* agg  = xw   + (long long)N16 * 64;
    float* hbuf = agg  + (long long)N16 * 64;

    // ---- degree (A + I) and per-edge norms ----
    k_init_deg <<<cdiv(N, TPB), TPB, 0, stream>>>(deg, N);
    k_deg_accum<<<cdiv(E, TPB), TPB, 0, stream>>>(dst, deg, E);
    k_deg_final<<<cdiv(N, TPB), TPB, 0, stream>>>(deg, dis, dinv, N);
    k_edge_norm<<<cdiv(E, TPB), TPB, 0, stream>>>(src, dst, dis, norm, E);

    // ---- padded copies (guard-free GEMMs) ----
    k_pad_x<<<cdiv((long long)N16 * 8, TPB), TPB, 0, stream>>>(x, xpad, N, N16);
    k_pad_w<<<cdiv(8 * 64, TPB),  TPB, 0, stream>>>(Wt[0], w0p, 7, 64, 8, 64);
    k_pad_w<<<cdiv(64 * 16, TPB), TPB, 0, stream>>>(Wt[5], w5p, 64, 7, 64, 16);

    const int nRowTiles = N16 / 16;
    const int gemmGrid  = cdiv(nRowTiles, TPB / 32);

    // Per-layer geometry: FINP (padded K / ldA), LD (padded fout), valid fout.
    // L0: 8->64  L1: 64->64  L2: 64->32  L3: 32->64  L4: 64->64  L5: 64->16(7)
    const float* hin = xpad;
    for (int i = 0; i < 6; ++i) {
        int ld, nfeat;
        switch (i) {
            case 0: ld = 64; nfeat = 64;
                k_gemm_wmma_f32<8, 4><<<gemmGrid, TPB, 0, stream>>>(hin, w0p,  xw, nRowTiles); break;
            case 1: ld = 64; nfeat = 64;
                k_gemm_wmma_f32<64, 4><<<gemmGrid, TPB, 0, stream>>>(hin, Wt[1], xw, nRowTiles); break;
            case 2: ld = 32; nfeat = 32;
                k_gemm_wmma_f32<64, 2><<<gemmGrid, TPB, 0, stream>>>(hin, Wt[2], xw, nRowTiles); break;
            case 3: ld = 64; nfeat = 64;
                k_gemm_wmma_f32<32, 4><<<gemmGrid, TPB, 0, stream>>>(hin, Wt[3], xw, nRowTiles); break;
            case 4: ld = 64; nfeat = 64;
                k_gemm_wmma_f32<64, 4><<<gemmGrid, TPB, 0, stream>>>(hin, Wt[4], xw, nRowTiles); break;
            default: ld = 16; nfeat = 7;   // padded cols 7..15 are zero
                k_gemm_wmma_f32<64, 1><<<gemmGrid, TPB, 0, stream>>>(hin, w5p,  xw, nRowTiles); break;
        }

        // agg = 0 over the padded slab
        const long long nslab = (long long)N16 * ld;
        k_zero<<<cdiv(nslab, TPB), TPB, 0, stream>>>(agg, nslab);

        // scatter: chunks covers >= nfeat features, 4*chunks <= ld, pow2
        int chunks = (i == 5) ? 2 : ld / 4;        // 16, 8 or 2
        int shift = 0; while ((1 << shift) < chunks) ++shift;
        const long long nthr = (long long)E * chunks;
        k_edge_scatter<<<cdiv(nthr, TPB), TPB, 0, stream>>>(
            xw, norm, src, dst, agg, E, ld, shift, chunks - 1);

        // h_next = relu(agg + dinv*xw + b); layer 2 mirrors latent to d_out,
        // layer 5 compacts to N x 7 at the front of d_out.
        float* o1 = (i == 5) ? out : hbuf;
        float* o2 = (i == 2) ? latent : nullptr;
        int ld_out = (i == 5) ? 7 : ld;
        const long long nc = (long long)N * nfeat;
        k_finalize<<<cdiv(nc, TPB), TPB, 0, stream>>>(
            agg, xw, dinv, bt[i], o1, o2, N, nfeat, ld, ld_out);

        hin = hbuf;
    }
}